// ConditionalSelfAttnEncoder_13340168421490
// MI455X (gfx1250) — compile-verified
//
#include <hip/hip_runtime.h>
#include <math.h>

// Problem dims (fixed by the reference)
#define HD 128   // hidden
#define LD 128   // doc_len
#define ND 32    // num docs
#define BD 4     // bsz
#define PITCH 132  // LDS row pitch: stride 4 mod 64 banks -> conflict-free column reads

typedef float v2f __attribute__((ext_vector_type(2)));
typedef float v8f __attribute__((ext_vector_type(8)));
typedef unsigned int u32x4 __attribute__((ext_vector_type(4)));
typedef int i32x4 __attribute__((ext_vector_type(4)));
typedef int i32x8 __attribute__((ext_vector_type(8)));

// V_WMMA_F32_16X16X4_F32 (codegen-confirmed round 1):
// D(16x16 f32, 8 VGPR) = A(16x4 f32, 2 VGPR) x B(4x16 f32, 2 VGPR) + C
__device__ __forceinline__ v8f wmma_f32(v2f a, v2f b, v8f c) {
  return __builtin_amdgcn_wmma_f32_16x16x4_f32(false, a, false, b, (short)0, c, false, false);
}

// ---------------------------------------------------------------------------
// TDM: DMA a 128x128 f32 tile (row stride `stride_elems` in memory) into LDS
// with hardware padding of 4 DWORDs after every 128 DWORDs -> LDS pitch 132.
// D# layout per CDNA5 ISA ch. 8 (group0 128b, group1 256b; groups 2/3 zero
// for a 2D tile). Issued once per wave; tracked by TENSORcnt.
// This toolchain exposes the 6-arg builtin (extra i32x8 descriptor words).
// ---------------------------------------------------------------------------
#if __has_builtin(__builtin_amdgcn_tensor_load_to_lds)
#define HAVE_TDM 1
__device__ __forceinline__ void tdm_load_128x128_f32(float* lds_dst,
                                                     const float* gsrc,
                                                     unsigned stride_elems) {
  unsigned long long ga = (unsigned long long)(uintptr_t)gsrc;
  u32x4 g0;
  g0[0] = 1u;                                   // count=1, user descriptor
  g0[1] = (unsigned)(uintptr_t)lds_dst;         // LDS byte address (low 32 bits)
  g0[2] = (unsigned)(ga & 0xFFFFFFFFu);         // global_addr[31:0]
  g0[3] = (unsigned)((ga >> 32) & 0x01FFFFFFu)  // global_addr[56:32]
        | (2u << 30);                           // type=2 ("image")
  i32x8 g1;
  g1[0] = (int)((2u << 16)        // data_size = 4B
              | (1u << 20)        // pad_enable
              | (6u << 22)        // pad_interval: every 128 DWORDs
              | (3u << 25));      // pad_amount: 4 DWORDs  -> pitch 132
  g1[1] = (int)(128u << 16);      // tensor_dim0 = 128 (bits [63:48] lo16)
  g1[2] = (int)(128u << 16);      // dim0 hi16 = 0 | tensor_dim1 = 128 (lo16)
  g1[3] = (int)(128u << 16);      // dim1 hi16 = 0 | tile_dim0 = 128
  g1[4] = (int)128u;              // tile_dim1 = 128 | tile_dim2 = 0
  g1[5] = (int)stride_elems;      // tensor_dim0_stride (data_size units) lo32
  g1[6] = 0;                      // stride hi16 | tensor_dim1_stride lo16
  g1[7] = 0;
  i32x4 z4 = {0, 0, 0, 0};
  i32x8 z8 = {0, 0, 0, 0, 0, 0, 0, 0};
  __builtin_amdgcn_tensor_load_to_lds(g0, g1, z4, z4, z8, 0);
}
#else
#define HAVE_TDM 0
#endif

// ---------------------------------------------------------------------------
// K1: ent_agg[n,h] = sum_l mask[n,l] * word_ent_info[n,l,h]
// ---------------------------------------------------------------------------
__global__ void ent_agg_kernel(const float* __restrict__ wei,
                               const int* __restrict__ wmask,
                               float* __restrict__ ent) {
  int tid = blockIdx.x * blockDim.x + threadIdx.x;   // 0 .. N*H-1
  int n = tid >> 7;
  int h = tid & 127;
  const float* wp = wei + (size_t)n * LD * HD + h;
  const int* mp = wmask + n * LD;
  float s = 0.f;
  #pragma unroll 8
  for (int l = 0; l < LD; ++l) s += (float)mp[l] * wp[l * HD];
  ent[tid] = s;
}

// ---------------------------------------------------------------------------
// K2: Bt[n,j,g] = sum_e doc[n,j,e]*ent[n,e]*W1b[g,e]
//     Ct[n,j,g] = sum_e doc[n,j,e]*q[n/8,e] *W1c[g,e]
// One block per (n,j); thread = g.
// ---------------------------------------------------------------------------
__global__ void btct_kernel(const float* __restrict__ doc,
                            const float* __restrict__ ent,
                            const float* __restrict__ q,
                            const float* __restrict__ W1,
                            float* __restrict__ Bt,
                            float* __restrict__ Ct) {
  __shared__ float de[HD];
  __shared__ float dq[HD];
  int nj = blockIdx.x;            // n*L + j
  int n  = nj >> 7;
  int g  = threadIdx.x;
  {
    int e = threadIdx.x;
    float d = doc[(size_t)nj * HD + e];
    de[e] = d * ent[n * HD + e];
    dq[e] = d * q[(n >> 3) * HD + e];   // N/B = 8 docs per batch row
  }
  __syncthreads();
  const float* wb = W1 + (size_t)g * (3 * HD) + HD;       // W1b row g
  const float* wc = W1 + (size_t)g * (3 * HD) + 2 * HD;   // W1c row g
  float bacc = 0.f, cacc = 0.f;
  #pragma unroll 8
  for (int e = 0; e < HD; ++e) {
    bacc += de[e] * wb[e];
    cacc += dq[e] * wc[e];
  }
  Bt[(size_t)nj * HD + g] = bacc;
  Ct[(size_t)nj * HD + g] = cacc;
}

// ---------------------------------------------------------------------------
// K3: score[n,j,k] for one (n,k) per workgroup.
//   A-tile GEMM: doc[n] (L x H) @ (W1a ⊙ doc[n,k,:])^T  -> [L x H]
//   fused: tanh( . + Bt[n,j,:] + Ct[n,k,:] + b1) · w2, scale, mask.
// 256 threads = 8 waves; wave w owns j-strip [16w, 16w+16).
// W1a kept in natural [g][e] layout (pitch 132), scaled in place by doc[n,k,:]
// so both WMMA operands are contiguous b64 LDS loads.
// ---------------------------------------------------------------------------
__global__ void score_kernel(const float* __restrict__ doc,
                             const int* __restrict__ dmask,
                             const float* __restrict__ W1,
                             const float* __restrict__ b1,
                             const float* __restrict__ w2,
                             const float* __restrict__ b2,
                             const float* __restrict__ Bt,
                             const float* __restrict__ Ct,
                             float* __restrict__ score) {
  extern __shared__ float smem[];
  float* doc_s  = smem;                    // [L][PITCH] doc rows   (A operand)
  float* w1a_s  = doc_s + LD * PITCH;      // [g][PITCH] W1a rows   (B operand, scaled)
  float* dock_s = w1a_s + HD * PITCH;      // [H] doc[n,k,:]
  float* ctk_s  = dock_s + HD;             // [H]
  float* b1_s   = ctk_s + HD;              // [H]
  float* w2_s   = b1_s + HD;               // [H]

  int n = blockIdx.x >> 7;
  int k = blockIdx.x & 127;
  int tid = threadIdx.x;
  const float* docn = doc + (size_t)n * LD * HD;

#if HAVE_TDM
  if (tid < 32) {   // wave 0 drives the Tensor Data Mover
    tdm_load_128x128_f32(doc_s, docn, HD);
    tdm_load_128x128_f32(w1a_s, W1, 3 * HD);
    __builtin_amdgcn_s_wait_tensorcnt(0);
  }
#else
  for (int i = tid; i < LD * HD; i += 256) {
    int r = i >> 7, c = i & 127;
    doc_s[r * PITCH + c] = docn[i];
    w1a_s[r * PITCH + c] = W1[(size_t)r * (3 * HD) + c];
  }
#endif
  if (tid < HD) {
    dock_s[tid] = docn[k * HD + tid];
    ctk_s[tid]  = Ct[((size_t)n * LD + k) * HD + tid];
    b1_s[tid]   = b1[tid];
    w2_s[tid]   = w2[tid];
  }
  __syncthreads();
  // Scale W1a rows by doc[n,k,:] in place: B[e][g] = W1a[g][e] * dock[e].
  for (int i = tid; i < HD * HD; i += 256) {
    int g = i >> 7, e = i & 127;
    w1a_s[g * PITCH + e] *= dock_s[e];
  }
  __syncthreads();

  int wid  = tid >> 5;
  int lane = tid & 31;
  int m    = lane & 15;     // A: row M, B: col N, C: col N
  int half = lane >> 4;     // K-pair select for A/B; +8 row offset for C
  int j0   = wid * 16;
  float b2v = b2[0];
  int dmk = dmask[n * LD + k];
  const float* Btn = Bt + (size_t)n * LD * HD;
  __builtin_prefetch(Btn + j0 * HD, 0, 0);   // global_prefetch_b8 for the Bt strip

  float p[8] = {0, 0, 0, 0, 0, 0, 0, 0};
  for (int gt = 0; gt < 8; ++gt) {
    int g0 = gt * 16;
    v8f acc = {0, 0, 0, 0, 0, 0, 0, 0};
    const float* arow = doc_s + (j0 + m) * PITCH + half * 2;   // A[m][e], e pair
    const float* brow = w1a_s + (g0 + m) * PITCH + half * 2;   // B[e][g0+m], e pair
    #pragma unroll 4
    for (int ks = 0; ks < 32; ++ks) {
      v2f a = *(const v2f*)(arow + ks * 4);   // ds_load_b64
      v2f b = *(const v2f*)(brow + ks * 4);   // ds_load_b64
      acc = wmma_f32(a, b, acc);
    }
    // Epilogue for this g-tile: C elem (M = v + 8*half, N = g0+m)
    int g = g0 + m;
    float ct = ctk_s[g], bb = b1_s[g], wg = w2_s[g];
    #pragma unroll
    for (int v = 0; v < 8; ++v) {
      int j = j0 + v + half * 8;
      float hv = tanhf(acc[v] + Btn[(size_t)j * HD + g] + ct + bb);
      p[v] += hv * wg;
    }
  }
  // Reduce across the 16 lanes of each half (masks < 16 stay inside the half).
  #pragma unroll
  for (int v = 0; v < 8; ++v) {
    float s = p[v];
    s += __shfl_xor(s, 8, 32);
    s += __shfl_xor(s, 4, 32);
    s += __shfl_xor(s, 2, 32);
    s += __shfl_xor(s, 1, 32);
    if (m == 0) {
      int j = j0 + v + half * 8;
      float sc = (s + b2v) * 0.08838834764831845f;   // 1/sqrt(128)
      if (dmask[n * LD + j] * dmk == 0) sc = -1e9f;
      score[((size_t)n * LD + j) * LD + k] = sc;
    }
  }
}

// ---------------------------------------------------------------------------
// K4: masked softmax over k per (n,j) row; write weights in place.
// ---------------------------------------------------------------------------
__global__ void softmax_kernel(const int* __restrict__ dmask,
                               float* __restrict__ score) {
  __shared__ float red[LD];
  int row = blockIdx.x;   // n*L + j
  int n = row >> 7, j = row & 127;
  int k = threadIdx.x;
  float s = score[(size_t)row * LD + k];
  red[k] = s; __syncthreads();
  for (int off = 64; off > 0; off >>= 1) {
    if (k < off) red[k] = fmaxf(red[k], red[k + off]);
    __syncthreads();
  }
  float mx = red[0]; __syncthreads();
  float e = expf(s - mx);
  red[k] = e; __syncthreads();
  for (int off = 64; off > 0; off >>= 1) {
    if (k < off) red[k] += red[k + off];
    __syncthreads();
  }
  float w = e / red[0];
  if (dmask[n * LD + j] * dmask[n * LD + k] == 0) w = 0.f;
  score[(size_t)row * LD + k] = w;
}

// ---------------------------------------------------------------------------
// K5: out[n] = w[n] (LxL) @ doc[n] (LxH), one WG per n, WMMA tiled.
// ---------------------------------------------------------------------------
__global__ void out_gemm_kernel(const float* __restrict__ doc,
                                const float* __restrict__ wmat,
                                float* __restrict__ out) {
  extern __shared__ float smem[];
  float* w_s = smem;                 // [L][PITCH]  A: weight rows j over k
  float* d_s = smem + LD * PITCH;    // [L][PITCH]  B: doc rows k over h
  int n = blockIdx.x, tid = threadIdx.x;
#if HAVE_TDM
  if (tid < 32) {
    tdm_load_128x128_f32(w_s, wmat + (size_t)n * LD * LD, LD);
    tdm_load_128x128_f32(d_s, doc + (size_t)n * LD * HD, HD);
    __builtin_amdgcn_s_wait_tensorcnt(0);
  }
#else
  for (int i = tid; i < LD * HD; i += 256) {
    int r = i >> 7, c = i & 127;
    w_s[r * PITCH + c] = wmat[(size_t)n * LD * LD + i];
    d_s[r * PITCH + c] = doc[(size_t)n * LD * HD + i];
  }
#endif
  __syncthreads();
  int wid = tid >> 5, lane = tid & 31;
  int m = lane & 15, half = lane >> 4;
  int j0 = wid * 16;
  for (int ht = 0; ht < 8; ++ht) {
    int h0 = ht * 16;
    v8f acc = {0, 0, 0, 0, 0, 0, 0, 0};
    const float* arow = w_s + (j0 + m) * PITCH + half * 2;
    #pragma unroll 4
    for (int ks = 0; ks < 32; ++ks) {
      int e = ks * 4 + half * 2;
      v2f a = *(const v2f*)(arow + ks * 4);   // ds_load_b64
      v2f b;
      b.x = d_s[e * PITCH + h0 + m];          // B[e][h] rows -> two b32 loads
      b.y = d_s[(e + 1) * PITCH + h0 + m];
      acc = wmma_f32(a, b, acc);
    }
    #pragma unroll
    for (int v = 0; v < 8; ++v)
      out[(size_t)n * LD * HD + (j0 + v + half * 8) * HD + h0 + m] = acc[v];
  }
}

// ---------------------------------------------------------------------------
extern "C" void kernel_launch(void* const* d_in, const int* in_sizes, int n_in,
                              void* d_out, int out_size, void* d_ws, size_t ws_size,
                              hipStream_t stream) {
  const float* wei   = (const float*)d_in[0];  // [N,L,H]
  const int*   wmask = (const int*)  d_in[1];  // [N,L]
  const float* doc   = (const float*)d_in[2];  // [N,L,H]
  const int*   dmask = (const int*)  d_in[3];  // [N,L]
  const float* q     = (const float*)d_in[4];  // [B,H]
  const float* W1    = (const float*)d_in[5];  // [H,3H]
  const float* b1    = (const float*)d_in[6];  // [H]
  const float* w2    = (const float*)d_in[7];  // [1,H]
  const float* b2    = (const float*)d_in[8];  // [1]
  float* out = (float*)d_out;                  // [N,L,H] fp32

  // Scratch layout (floats): ent 4096 | Bt 524288 | Ct 524288 | score/w 524288
  // Total ~6.3 MB.
  float* ws    = (float*)d_ws;
  float* ent   = ws;
  float* Bt    = ent + ND * HD;
  float* Ct    = Bt + (size_t)ND * LD * HD;
  float* score = Ct + (size_t)ND * LD * HD;

  const int SMEM2 = (2 * LD * PITCH + 4 * HD) * (int)sizeof(float);  // ~134 KB
  const int SMEM4 = (2 * LD * PITCH) * (int)sizeof(float);           // ~132 KB
  (void)hipFuncSetAttribute(reinterpret_cast<const void*>(&score_kernel),
                            hipFuncAttributeMaxDynamicSharedMemorySize, SMEM2);
  (void)hipFuncSetAttribute(reinterpret_cast<const void*>(&out_gemm_kernel),
                            hipFuncAttributeMaxDynamicSharedMemorySize, SMEM4);

  ent_agg_kernel<<<(ND * HD) / 256, 256, 0, stream>>>(wei, wmask, ent);
  btct_kernel<<<ND * LD, HD, 0, stream>>>(doc, ent, q, W1, Bt, Ct);
  score_kernel<<<ND * LD, 256, SMEM2, stream>>>(doc, dmask, W1, b1, w2, b2, Bt, Ct, score);
  softmax_kernel<<<ND * LD, LD, 0, stream>>>(dmask, score);
  out_gemm_kernel<<<ND, 256, SMEM4, stream>>>(doc, score, out);
}